// MPNN_9216999817569
// MI455X (gfx1250) — compile-verified
//
#include <hip/hip_runtime.h>
#include <math.h>

#define NN 20000
#define EE 320000
#define DD 128
#define HH 8
#define GG 64
#define CUT 5.0f
#define LNE 1e-5f

// padded LDS strides (floats) for bank-conflict-free WMMA fragment reads
#define LDA 132   // A tiles: bank = 4*m + k  -> 16 distinct banks per half-wave
#define LDB 136   // B tiles: half-wave row groups land in disjoint 16-bank windows
#define LDAF 68   // filter A tile (K=64)

typedef float v2f __attribute__((ext_vector_type(2)));
typedef float v8f __attribute__((ext_vector_type(8)));
typedef int vi4 __attribute__((vector_size(16)));

// ---------- gfx1250 async global->LDS staging (guarded) ----------
#if defined(__has_builtin)
#if __has_builtin(__builtin_amdgcn_global_load_async_to_lds_b128) && \
    __has_builtin(__builtin_amdgcn_s_wait_asynccnt)
#define USE_ASYNC 1
#endif
#endif
#ifndef USE_ASYNC
#define USE_ASYNC 0
#endif

__device__ __forceinline__ void copy_f4_to_lds(const float* g, float* l) {
#if USE_ASYNC
    __builtin_amdgcn_global_load_async_to_lds_b128(
        (__attribute__((address_space(1))) vi4*)g,
        (__attribute__((address_space(3))) vi4*)l, 0, 0);
#else
    *(float4*)l = *(const float4*)g;
#endif
}
__device__ __forceinline__ void lds_copy_wait() {
#if USE_ASYNC
    __builtin_amdgcn_s_wait_asynccnt(0);
#endif
}

// stage a row-major [128][128] weight matrix into LDS with row stride LDB
__device__ __forceinline__ void stage_weight(float* s_w, const float* g_w,
                                             int tid, int nthr) {
    for (int j = tid; j < (DD * DD) / 4; j += nthr) {
        int row = j >> 5, c4 = (j << 2) & (DD - 1);
        copy_f4_to_lds(g_w + (size_t)row * DD + c4, s_w + row * LDB + c4);
    }
}

// ---------- helpers ----------
__device__ __forceinline__ unsigned fenc(float f) {
    unsigned u = __float_as_uint(f);
    return u ^ ((unsigned)((int)u >> 31) | 0x80000000u);
}
__device__ __forceinline__ float fdec(unsigned e) {
    unsigned u = (e & 0x80000000u) ? (e ^ 0x80000000u) : ~e;
    return __uint_as_float(u);
}

// One 16x16 fp32 output tile via V_WMMA_F32_16X16X4_F32.
// A: row-major [16][lda] (LDS). B: row-major [K][ldb] (LDS).
template <int K>
__device__ __forceinline__ v8f tile_gemm(const float* __restrict__ sA, int lda,
                                         const float* __restrict__ B, int ldb,
                                         int ntile, int lane) {
    v8f acc = {0.f, 0.f, 0.f, 0.f, 0.f, 0.f, 0.f, 0.f};
    const int m   = lane & 15;          // A row (M)
    const int kh  = (lane >> 4) << 1;   // lanes 0-15 -> K0/K1, 16-31 -> K2/K3
    const int col = (ntile << 4) + (lane & 15);
#pragma unroll
    for (int kk = 0; kk < K; kk += 4) {
        v2f A, Bv;
        A.x  = sA[m * lda + kk + kh];
        A.y  = sA[m * lda + kk + kh + 1];
        Bv.x = B[(kk + kh) * ldb + col];
        Bv.y = B[(kk + kh + 1) * ldb + col];
        acc = __builtin_amdgcn_wmma_f32_16x16x4_f32(false, A, false, Bv,
                                                    (short)0, acc, false, false);
    }
    return acc;
}

// ---------- K0: zero accumulators ----------
__global__ void k_zero(float* p, int n) {
    int i = blockIdx.x * 256 + threadIdx.x;
    if (i < n) p[i] = 0.0f;
}

// ---------- K1: RBF + filter GEMM (W out, with cutoff) ----------
__global__ __launch_bounds__(128) void k_filter(const float* __restrict__ pos,
                                                const float* __restrict__ fw,
                                                const float* __restrict__ fb,
                                                const int* __restrict__ src,
                                                const int* __restrict__ dst,
                                                float* __restrict__ Wout,
                                                float* __restrict__ Cws) {
    __shared__ float s_fw[GG * LDB];        // padded weights  ~34.8 KB
    __shared__ float s_win[4][16 * LDAF];   // padded A tiles  ~17.4 KB
    __shared__ float s_r[4][16];
    __shared__ float s_c[4][16];
    const int tid = threadIdx.x, w = tid >> 5, lane = tid & 31;
    for (int j = tid; j < (GG * DD) / 4; j += 128) {
        int row = j >> 5, c4 = (j << 2) & (DD - 1);
        copy_f4_to_lds(fw + (size_t)row * DD + c4, s_fw + row * LDB + c4);
    }
    lds_copy_wait();
    const int e0 = (blockIdx.x * 4 + w) * 16;
    if (lane < 16) {
        int e = e0 + lane;
        int s = src[e], d = dst[e];
        float dx = pos[d * 3 + 0] - pos[s * 3 + 0];
        float dy = pos[d * 3 + 1] - pos[s * 3 + 1];
        float dz = pos[d * 3 + 2] - pos[s * 3 + 2];
        float r = sqrtf(dx * dx + dy * dy + dz * dz);
        float C = (r < CUT) ? 0.5f * (cosf(r * (3.14159265358979f / CUT)) + 1.0f) : 0.0f;
        s_r[w][lane] = r;
        s_c[w][lane] = C;
        Cws[e] = C;
    }
    __syncthreads();
    const float width = CUT / (float)(GG - 1);
    const float coef  = -0.5f / (width * width);
    for (int i = lane; i < 16 * GG; i += 32) {
        int el = i >> 6, g = i & 63;
        float diff = s_r[w][el] - width * (float)g;
        s_win[w][el * LDAF + g] = 2.0f * __expf(coef * diff * diff) - 1.0f;
    }
    __syncthreads();
    const int hg = lane >> 4;
    for (int n = 0; n < 8; ++n) {
        v8f acc = tile_gemm<GG>(s_win[w], LDAF, s_fw, LDB, n, lane);
        int col = n * 16 + (lane & 15);
        float fbv = fb[col];
#pragma unroll
        for (int v = 0; v < 8; ++v) {
            int row = v + (hg << 3);
            Wout[(size_t)(e0 + row) * DD + col] = (acc[v] + fbv) * s_c[w][row];
        }
    }
}

// ---------- K2: q = W*h1[dst] + t, LayerNorm (1 wave / edge) ----------
__global__ __launch_bounds__(256) void k_qln(const float* __restrict__ Wout,
                                             const float* __restrict__ h1,
                                             const float* __restrict__ t,
                                             const float* __restrict__ ln_g,
                                             const float* __restrict__ ln_b,
                                             const int* __restrict__ dst,
                                             float* __restrict__ qws) {
    const int w = threadIdx.x >> 5, lane = threadIdx.x & 31;
    const int e = blockIdx.x * 8 + w;
    const float4* Wr = (const float4*)(Wout + (size_t)e * DD);
    const float4* hr = (const float4*)(h1 + (size_t)dst[e] * DD);
    const float4* tr = (const float4*)(t + (size_t)e * DD);
    float4 a = Wr[lane], b = hr[lane], c = tr[lane];
    float q0 = a.x * b.x + c.x, q1 = a.y * b.y + c.y;
    float q2 = a.z * b.z + c.z, q3 = a.w * b.w + c.w;
    float sm = q0 + q1 + q2 + q3;
    for (int mk = 16; mk; mk >>= 1) sm += __shfl_xor(sm, mk, 32);
    float mu = sm * (1.0f / DD);
    float d0 = q0 - mu, d1 = q1 - mu, d2 = q2 - mu, d3 = q3 - mu;
    float vs = d0 * d0 + d1 * d1 + d2 * d2 + d3 * d3;
    for (int mk = 16; mk; mk >>= 1) vs += __shfl_xor(vs, mk, 32);
    float inv = rsqrtf(vs * (1.0f / DD) + LNE);
    const float4* gg = (const float4*)ln_g;
    const float4* bb = (const float4*)ln_b;
    float4 gv = gg[lane], bv = bb[lane];
    float4 o;
    o.x = d0 * inv * gv.x + bv.x;
    o.y = d1 * inv * gv.y + bv.y;
    o.z = d2 * inv * gv.z + bv.z;
    o.w = d3 * inv * gv.w + bv.w;
    ((float4*)(qws + (size_t)e * DD))[lane] = o;
}

// ---------- K3: q_vec = q[q_id] @ Wq ----------
__global__ __launch_bounds__(64) void k_qvec(const float* __restrict__ qws,
                                             const float* __restrict__ Wq,
                                             const int* __restrict__ q_id,
                                             float* __restrict__ qvec) {
    __shared__ float s_w[DD * LDB];       // ~69.6 KB
    __shared__ float s_a[2][16 * LDA];    // ~16.9 KB
    __shared__ int s_qid[2][16];
    const int tid = threadIdx.x, w = tid >> 5, lane = tid & 31;
    const int n0 = (blockIdx.x * 2 + w) * 16;
    stage_weight(s_w, Wq, tid, 64);
    if (lane < 16) s_qid[w][lane] = q_id[n0 + lane];
    __syncthreads();
    for (int j = lane; j < (16 * DD) / 4; j += 32) {
        int row = j >> 5, c4 = (j << 2) & (DD - 1);
        copy_f4_to_lds(qws + (size_t)s_qid[w][row] * DD + c4,
                       &s_a[w][row * LDA + c4]);
    }
    lds_copy_wait();
    __syncthreads();
    const int hg = lane >> 4;
    for (int n = 0; n < 8; ++n) {
        v8f acc = tile_gemm<DD>(s_a[w], LDA, s_w, LDB, n, lane);
        int col = n * 16 + (lane & 15);
#pragma unroll
        for (int v = 0; v < 8; ++v)
            qvec[(size_t)(n0 + v + (hg << 3)) * DD + col] = acc[v];
    }
}

// ---------- K4: k = q@Wk fused with per-head logits + segment max; v = q@Wv ----------
__global__ __launch_bounds__(128) void k_kv(const float* __restrict__ qws,
                                            const float* __restrict__ Wk,
                                            const float* __restrict__ Wv,
                                            const float* __restrict__ qvec,
                                            const int* __restrict__ src,
                                            float* __restrict__ vvec,
                                            float* __restrict__ logit,
                                            unsigned* __restrict__ menc) {
    __shared__ float s_w[DD * LDB];       // ~69.6 KB, Wk then Wv
    __shared__ float s_a[4][16 * LDA];    // ~33.8 KB q tiles
    __shared__ int s_src[4][16];
    const int tid = threadIdx.x, w = tid >> 5, lane = tid & 31;
    const int e0 = (blockIdx.x * 4 + w) * 16;
    stage_weight(s_w, Wk, tid, 128);
    if (lane < 16) s_src[w][lane] = src[e0 + lane];
    for (int j = lane; j < (16 * DD) / 4; j += 32) {
        int row = j >> 5, c4 = (j << 2) & (DD - 1);
        copy_f4_to_lds(qws + (size_t)(e0 + row) * DD + c4,
                       &s_a[w][row * LDA + c4]);
    }
    lds_copy_wait();
    __syncthreads();
    const int hg = lane >> 4;
    // head h == N-tile h (P=16): k tile, dot with q_vec[src], 16-lane reduce
    for (int h = 0; h < HH; ++h) {
        v8f acc = tile_gemm<DD>(s_a[w], LDA, s_w, LDB, h, lane);
        int col = h * 16 + (lane & 15);
#pragma unroll
        for (int v = 0; v < 8; ++v) {
            int row = v + (hg << 3);
            float p = acc[v] * qvec[(size_t)s_src[w][row] * DD + col];
            p += __shfl_xor(p, 1, 32);
            p += __shfl_xor(p, 2, 32);
            p += __shfl_xor(p, 4, 32);
            p += __shfl_xor(p, 8, 32);
            if ((lane & 15) == 0) {
                float lg = p * 0.25f;   // 1/sqrt(P), P=16
                logit[(size_t)(e0 + row) * HH + h] = lg;
                atomicMax(&menc[(size_t)s_src[w][row] * HH + h], fenc(lg));
            }
        }
    }
    __syncthreads();                       // all waves done reading Wk
    stage_weight(s_w, Wv, tid, 128);
    lds_copy_wait();
    __syncthreads();
    for (int n = 0; n < 8; ++n) {
        v8f acc = tile_gemm<DD>(s_a[w], LDA, s_w, LDB, n, lane);
        int col = n * 16 + (lane & 15);
#pragma unroll
        for (int v = 0; v < 8; ++v)
            vvec[(size_t)(e0 + v + (hg << 3)) * DD + col] = acc[v];
    }
}

// ---------- K5: a = exp(logit - m[src]), z += a ----------
__global__ void k_exp(float* __restrict__ lg, const unsigned* __restrict__ menc,
                      const int* __restrict__ src, float* __restrict__ z) {
    int i = blockIdx.x * 256 + threadIdx.x;
    if (i >= EE * HH) return;
    int e = i >> 3, h = i & 7;
    int sidx = src[e] * HH + h;
    float a = __expf(lg[i] - fdec(menc[sidx]));
    lg[i] = a;
    atomicAdd(&z[sidx], a);
}

// ---------- K6: att = a/z * C, s += att ----------
__global__ void k_att(float* __restrict__ lg, const float* __restrict__ z,
                      const float* __restrict__ Cws, const int* __restrict__ src,
                      float* __restrict__ s) {
    int i = blockIdx.x * 256 + threadIdx.x;
    if (i >= EE * HH) return;
    int e = i >> 3, h = i & 7;
    int sidx = src[e] * HH + h;
    float att = lg[i] / z[sidx] * Cws[e];
    lg[i] = att;
    atomicAdd(&s[sidx], att);
}

// ---------- K7: h2[src] += v * att/s ----------
__global__ void k_scatter(const float* __restrict__ vv, const float* __restrict__ lg,
                          const float* __restrict__ s, const int* __restrict__ src,
                          float* __restrict__ h2) {
    int i = blockIdx.x * 256 + threadIdx.x;
    if (i >= EE * DD) return;
    int e = i >> 7, c = i & 127, h = c >> 4;
    int sn = src[e];
    float attn = lg[(size_t)e * HH + h] / s[(size_t)sn * HH + h];
    atomicAdd(&h2[(size_t)sn * DD + c], vv[i] * attn);
}

// ---------- K8: m_agg = h2 @ Wo ----------
__global__ __launch_bounds__(64) void k_out(const float* __restrict__ h2,
                                            const float* __restrict__ Wo,
                                            float* __restrict__ magg) {
    __shared__ float s_w[DD * LDB];
    __shared__ float s_a[2][16 * LDA];
    const int tid = threadIdx.x, w = tid >> 5, lane = tid & 31;
    const int n0 = (blockIdx.x * 2 + w) * 16;
    stage_weight(s_w, Wo, tid, 64);
    for (int j = lane; j < (16 * DD) / 4; j += 32) {
        int row = j >> 5, c4 = (j << 2) & (DD - 1);
        copy_f4_to_lds(h2 + (size_t)(n0 + row) * DD + c4,
                       &s_a[w][row * LDA + c4]);
    }
    lds_copy_wait();
    __syncthreads();
    const int hg = lane >> 4;
    for (int n = 0; n < 8; ++n) {
        v8f acc = tile_gemm<DD>(s_a[w], LDA, s_w, LDB, n, lane);
        int col = n * 16 + (lane & 15);
#pragma unroll
        for (int v = 0; v < 8; ++v)
            magg[(size_t)(n0 + v + (hg << 3)) * DD + col] = acc[v];
    }
}

extern "C" void kernel_launch(void* const* d_in, const int* in_sizes, int n_in,
                              void* d_out, int out_size, void* d_ws, size_t ws_size,
                              hipStream_t stream) {
    (void)in_sizes; (void)n_in; (void)out_size; (void)ws_size;
    const float* pos  = (const float*)d_in[0];
    const float* h1   = (const float*)d_in[1];
    const float* t    = (const float*)d_in[2];
    const float* fw   = (const float*)d_in[3];
    const float* fb   = (const float*)d_in[4];
    const float* ln_g = (const float*)d_in[5];
    const float* ln_b = (const float*)d_in[6];
    const float* Wq   = (const float*)d_in[7];
    const float* Wk   = (const float*)d_in[8];
    const float* Wv   = (const float*)d_in[9];
    const float* Wo   = (const float*)d_in[10];
    const int* src    = (const int*)d_in[11];
    const int* dst    = (const int*)d_in[12];
    const int* q_id   = (const int*)d_in[13];

    float* magg = (float*)d_out;                       // [N,D]
    float* Wout = (float*)d_out + (size_t)NN * DD;     // [E,D]

    // workspace layout (floats)
    float* ws = (float*)d_ws;
    size_t offC  = 0;
    size_t offQ  = offC + EE;                    // q (post-LN)  [E,D]
    size_t offV  = offQ + (size_t)EE * DD;       // v_vec        [E,D]
    size_t offQV = offV + (size_t)EE * DD;       // q_vec        [N,D]
    size_t offL  = offQV + (size_t)NN * DD;      // logit/a/att  [E,H]
    size_t offM  = offL + (size_t)EE * HH;       // m_enc (u32)  [N,H]
    size_t offZ  = offM + (size_t)NN * HH;       // z            [N,H]
    size_t offS  = offZ + (size_t)NN * HH;       // s            [N,H]
    size_t offH2 = offS + (size_t)NN * HH;       // h2           [N,D]
    float* Cws = ws + offC;
    float* qws = ws + offQ;
    float* vv  = ws + offV;
    float* qv  = ws + offQV;
    float* lg  = ws + offL;
    unsigned* menc = (unsigned*)(ws + offM);
    float* z = ws + offZ;
    float* s = ws + offS;
    float* h2 = ws + offH2;

    int zn = 3 * NN * HH + NN * DD;  // menc, z, s, h2 (enc key 0 is the minimum)
    k_zero<<<(zn + 255) / 256, 256, 0, stream>>>(ws + offM, zn);
    k_filter<<<EE / 64, 128, 0, stream>>>(pos, fw, fb, src, dst, Wout, Cws);
    k_qln<<<EE / 8, 256, 0, stream>>>(Wout, h1, t, ln_g, ln_b, dst, qws);
    k_qvec<<<NN / 32, 64, 0, stream>>>(qws, Wq, q_id, qv);
    k_kv<<<EE / 64, 128, 0, stream>>>(qws, Wk, Wv, qv, src, vv, lg, menc);
    k_exp<<<(EE * HH + 255) / 256, 256, 0, stream>>>(lg, menc, src, z);
    k_att<<<(EE * HH + 255) / 256, 256, 0, stream>>>(lg, z, Cws, src, s);
    k_scatter<<<(EE * DD + 255) / 256, 256, 0, stream>>>(vv, lg, s, src, h2);
    k_out<<<NN / 32, 64, 0, stream>>>(h2, Wo, magg);
}